// ComposerGCN_22565758173359
// MI455X (gfx1250) — compile-verified
//
#include <hip/hip_runtime.h>

// ---------------------------------------------------------------------------
// GraphConv x2 for MI455X (gfx1250, wave32).
//   h   = relu( scatter_add(x[src],dst) @ Wrel1^T + x @ Wroot1^T + b1 )
//   out =       scatter_add(h[src],dst) @ Wrel2^T + h @ Wroot2^T + b2
// Memory-bound (~3.7 GB @ 23.3 TB/s). GEMMs use V_WMMA_F32_16X16X4_F32.
// ---------------------------------------------------------------------------

typedef float v2f __attribute__((ext_vector_type(2)));
typedef float v8f __attribute__((ext_vector_type(8)));

#define GC_N_NODES 100000
#define GC_F 128   // feature width of both aggregation passes (input & hidden)

// ---------------------------------------------------------------- zero fill
__global__ void gc_zero_kernel(float* __restrict__ p, int n) {
  int i = blockIdx.x * blockDim.x + threadIdx.x;
  if (i < n) p[i] = 0.0f;
}

// ---------------------------------------------------- edge scatter-add (f32)
// One wave per edge; lane l owns features [4l, 4l+4). The 51 MB agg array
// lives in the 192 MB L2, so global_atomic_add_f32 resolves in L2.
__global__ void gc_scatter_add_kernel(const float* __restrict__ feat,  // [N,128]
                                      const int* __restrict__ src,     // [E]
                                      const int* __restrict__ dst,     // [E]
                                      float* __restrict__ agg,         // [N,128]
                                      int n_edges) {
  int tid = blockIdx.x * blockDim.x + threadIdx.x;
  int e = tid >> 5;
  int lane = tid & 31;
  if (e >= n_edges) return;
  int s = src[e];
  int d = dst[e];
  const float4 v = *(const float4*)(feat + (size_t)s * GC_F + lane * 4);
  float* o = agg + (size_t)d * GC_F + lane * 4;
  atomicAdd(o + 0, v.x);
  atomicAdd(o + 1, v.y);
  atomicAdd(o + 2, v.z);
  atomicAdd(o + 3, v.w);
}

// --------------------------------------------- fused dual-GEMM + bias (+relu)
// out[M,NOUT] = agg @ w_rel^T + x @ w_root^T + bias, K = 128.
// One wave computes one 16x16 output tile with V_WMMA_F32_16X16X4_F32,
// chaining 2 WMMAs (rel + root) per K-step of 4 into one v8f accumulator.
//
// Operand layouts (wave32, ISA 7.12.2):
//   A 16x4 f32 : lane 0-15 -> M=lane, VGPR{0,1}=K{0,1}; lane 16-31 -> K{2,3}
//   B 4x16 f32 : lane 0-15 -> N=lane, VGPR{0,1}=K{0,1}; lane 16-31 -> K{2,3}
//   C/D 16x16  : VGPR i -> M = i + 8*(lane>>4), N = lane&15
template <int NOUT, bool RELU>
__global__ void gc_fused_gemm_kernel(const float* __restrict__ agg,    // [N,128]
                                     const float* __restrict__ x,      // [N,128]
                                     const float* __restrict__ w_rel,  // [NOUT,128]
                                     const float* __restrict__ w_root, // [NOUT,128]
                                     const float* __restrict__ bias,   // [NOUT]
                                     float* __restrict__ out,          // [N,NOUT]
                                     int n_nodes) {
  constexpr int K = GC_F;
  constexpr int nColTiles = NOUT / 16;

  const int wave = (blockIdx.x * blockDim.x + threadIdx.x) >> 5;
  const int lane = threadIdx.x & 31;
  const int rowTile = wave / nColTiles;
  const int colTile = wave % nColTiles;
  const int row0 = rowTile * 16;
  if (row0 >= n_nodes) return;
  const int col0 = colTile * 16;

  const int mr    = lane & 15;          // A-row / B-col within tile
  const int khalf = (lane >> 4) * 2;    // K sub-offset within 4-wide step

  const float* aggRow   = agg    + (size_t)(row0 + mr) * K + khalf;
  const float* xRow     = x      + (size_t)(row0 + mr) * K + khalf;
  // B(k, n) = W[n, k]  (W stored row-major [NOUT, K]) -> read W rows as cols
  const float* wRelCol  = w_rel  + (size_t)(col0 + mr) * K + khalf;
  const float* wRootCol = w_root + (size_t)(col0 + mr) * K + khalf;

  v8f c = {0.f, 0.f, 0.f, 0.f, 0.f, 0.f, 0.f, 0.f};

#pragma unroll 4
  for (int k = 0; k < K; k += 4) {
    v2f a0 = *(const v2f*)(aggRow + k);
    v2f b0 = *(const v2f*)(wRelCol + k);
    c = __builtin_amdgcn_wmma_f32_16x16x4_f32(false, a0, false, b0,
                                              (short)0, c, false, false);
    v2f a1 = *(const v2f*)(xRow + k);
    v2f b1 = *(const v2f*)(wRootCol + k);
    c = __builtin_amdgcn_wmma_f32_16x16x4_f32(false, a1, false, b1,
                                              (short)0, c, false, false);
  }

  const int n = lane & 15;
  const int mbase = 8 * (lane >> 4);
  const float bv = bias[col0 + n];
#pragma unroll
  for (int i = 0; i < 8; ++i) {
    float v = c[i] + bv;
    if (RELU) v = fmaxf(v, 0.0f);
    out[(size_t)(row0 + mbase + i) * NOUT + col0 + n] = v;
  }
}

// ---------------------------------------------------------------------------
extern "C" void kernel_launch(void* const* d_in, const int* in_sizes, int n_in,
                              void* d_out, int out_size, void* d_ws, size_t ws_size,
                              hipStream_t stream) {
  const float* x       = (const float*)d_in[0];
  const int*   edge    = (const int*)d_in[1];       // [2, E]
  const float* w_rel1  = (const float*)d_in[2];     // [128,128]
  const float* b_rel1  = (const float*)d_in[3];     // [128]
  const float* w_root1 = (const float*)d_in[4];     // [128,128]
  const float* w_rel2  = (const float*)d_in[5];     // [64,128]
  const float* b_rel2  = (const float*)d_in[6];     // [64]
  const float* w_root2 = (const float*)d_in[7];     // [64,128]
  float* out = (float*)d_out;

  const int n_edges = in_sizes[1] / 2;
  const int* src = edge;             // edge_index[0]
  const int* dst = edge + n_edges;   // edge_index[1]

  // workspace: agg [N,128] fp32, h [N,128] fp32
  float* agg = (float*)d_ws;
  float* h   = agg + (size_t)GC_N_NODES * GC_F;

  const int nfeat = GC_N_NODES * GC_F;
  const int zeroBlocks = (nfeat + 255) / 256;
  const int scatBlocks = (int)(((long long)n_edges * 32 + 255) / 256);

  // ---- layer 1 ----
  gc_zero_kernel<<<zeroBlocks, 256, 0, stream>>>(agg, nfeat);
  gc_scatter_add_kernel<<<scatBlocks, 256, 0, stream>>>(x, src, dst, agg, n_edges);
  {
    const int waves = ((GC_N_NODES + 15) / 16) * (128 / 16);
    const int blocks = (waves + 7) / 8;   // 8 wave32 per 256-thread block
    gc_fused_gemm_kernel<128, true><<<blocks, 256, 0, stream>>>(
        agg, x, w_rel1, w_root1, b_rel1, h, GC_N_NODES);
  }

  // ---- layer 2 ----
  gc_zero_kernel<<<zeroBlocks, 256, 0, stream>>>(agg, nfeat);
  gc_scatter_add_kernel<<<scatBlocks, 256, 0, stream>>>(h, src, dst, agg, n_edges);
  {
    const int waves = ((GC_N_NODES + 15) / 16) * (64 / 16);
    const int blocks = (waves + 7) / 8;
    gc_fused_gemm_kernel<64, false><<<blocks, 256, 0, stream>>>(
        agg, h, w_rel2, w_root2, b_rel2, out, GC_N_NODES);
  }
}